// GCNLayer_2929167695897
// MI455X (gfx1250) — compile-verified
//
#include <hip/hip_runtime.h>

typedef __attribute__((ext_vector_type(2))) float v2f;
typedef __attribute__((ext_vector_type(4))) float v4f;
typedef __attribute__((ext_vector_type(8))) float v8f;

#define FDIM 64

// ---------------------------------------------------------------------------
// Kernel 1: zero the accumulator (d_out doubles as h).
// ---------------------------------------------------------------------------
__global__ void gcn_zero_kernel(float* __restrict__ out, int n4) {
    int i = blockIdx.x * blockDim.x + threadIdx.x;
    if (i < n4) {
        v4f z = {0.0f, 0.0f, 0.0f, 0.0f};
        *(v4f*)(out + 4 * (size_t)i) = z;
    }
}

// ---------------------------------------------------------------------------
// Kernel 2: edge scatter-add.  h[dst[e], :] += feature[src[e], :]
// Each thread owns (edge, 4-feature chunk): one B128 load + 4 f32 atomics.
// 16 threads cooperate on one edge; src/dst loads broadcast out of L0.
// ---------------------------------------------------------------------------
__global__ void gcn_scatter_kernel(const float* __restrict__ feat,
                                   const int* __restrict__ src,
                                   const int* __restrict__ dst,
                                   float* __restrict__ h,
                                   int nedges) {
    int t = blockIdx.x * blockDim.x + threadIdx.x;
    int e = t >> 4;                 // edge index
    int q = (t & 15) << 2;          // feature base: 0,4,...,60
    if (e >= nedges) return;
    int s = src[e];
    int d = dst[e];
    v4f v = *(const v4f*)(feat + (size_t)s * FDIM + q);
    float* p = h + (size_t)d * FDIM + q;
    atomicAdd(p + 0, v.x);
    atomicAdd(p + 1, v.y);
    atomicAdd(p + 2, v.z);
    atomicAdd(p + 3, v.w);
}

// ---------------------------------------------------------------------------
// Kernel 3: in-place linear  out = h @ W^T + b  using V_WMMA_F32_16X16X4_F32.
// One wave -> one 16-row x 64-col output tile (4 accumulators, 16 K-steps).
//
// A (16x4 f32) layout: lane<16: v0=A[lane][k], v1=A[lane][k+1];
//                      lane>=16: v0=A[lane-16][k+2], v1=A[lane-16][k+3].
// B (4x16 f32) mirrors with N on lanes; B[k][n] = W[n][k] so each lane
// loads a contiguous float2 from W's row (ct*16 + lane&15).
// C/D (16x16 f32): lane half gives N=lane&15, VGPR r gives M=r+8*(lane>>4).
// In-place safety: the full 16x64 A block is loaded into VGPRs before any
// WMMA or store; each wave owns a disjoint row block.
// ---------------------------------------------------------------------------
__global__ void gcn_linear_wmma_kernel(float* __restrict__ hout,  // [N,64], in: h, out: result
                                       const float* __restrict__ W,   // [64,64] row-major [out,in]
                                       const float* __restrict__ bias,
                                       int nrowblocks) {
    const int lane = threadIdx.x & 31;
    const int wave = (blockIdx.x * (blockDim.x >> 5)) + (threadIdx.x >> 5);
    if (wave >= nrowblocks) return;   // whole-wave uniform; EXEC stays all-ones

    const int m    = lane & 15;
    const int half = lane >> 4;       // 0: K=0,1  1: K=2,3

    // ---- load full A block (16 rows x 64 cols) for this wave ----
    const float* arow = hout + ((size_t)(wave * 16 + m)) * FDIM + 2 * half;
    v2f afrag[16];
#pragma unroll
    for (int ks = 0; ks < 16; ++ks)
        afrag[ks] = *(const v2f*)(arow + ks * 4);

    // ---- accumulators initialized with the bias (depends on N = ct*16+m only) ----
    v8f acc[4];
#pragma unroll
    for (int ct = 0; ct < 4; ++ct) {
        float bb = bias[ct * 16 + m];
        v8f c = {bb, bb, bb, bb, bb, bb, bb, bb};
        acc[ct] = c;
    }

    // ---- 4 column tiles x 16 K-steps of f32 WMMA ----
#pragma unroll
    for (int ct = 0; ct < 4; ++ct) {
        const float* wrow = W + ((size_t)(ct * 16 + m)) * FDIM + 2 * half;
#pragma unroll
        for (int ks = 0; ks < 16; ++ks) {
            v2f bfrag = *(const v2f*)(wrow + ks * 4);
            acc[ct] = __builtin_amdgcn_wmma_f32_16x16x4_f32(
                /*neg_a=*/false, afrag[ks],
                /*neg_b=*/false, bfrag,
                /*c_mod=*/(short)0, acc[ct],
                /*reuse_a=*/false, /*reuse_b=*/false);
        }
    }

    // ---- store 16x64 result tile in-place ----
#pragma unroll
    for (int ct = 0; ct < 4; ++ct) {
#pragma unroll
        for (int r = 0; r < 8; ++r) {
            int row = wave * 16 + r + 8 * half;
            hout[(size_t)row * FDIM + ct * 16 + m] = acc[ct][r];
        }
    }
}

// ---------------------------------------------------------------------------
// Host-side launcher
// ---------------------------------------------------------------------------
extern "C" void kernel_launch(void* const* d_in, const int* in_sizes, int n_in,
                              void* d_out, int out_size, void* d_ws, size_t ws_size,
                              hipStream_t stream) {
    const float* feature = (const float*)d_in[0];   // [100000, 64] f32
    const int*   src     = (const int*)d_in[1];     // [1600000]
    const int*   dst     = (const int*)d_in[2];     // [1600000]
    const float* W       = (const float*)d_in[3];   // [64, 64] f32
    const float* bias    = (const float*)d_in[4];   // [64] f32

    float* out = (float*)d_out;                     // [100000, 64] f32, also h accumulator

    const int nnodes = in_sizes[0] / FDIM;          // 100000
    const int nedges = in_sizes[1];                 // 1600000

    // 1) zero the accumulator
    {
        int n4 = (nnodes * FDIM) / 4;
        int blocks = (n4 + 255) / 256;
        gcn_zero_kernel<<<blocks, 256, 0, stream>>>(out, n4);
    }

    // 2) scatter-add edges into d_out
    {
        long long threads = (long long)nedges * 16;
        int blocks = (int)((threads + 255) / 256);
        gcn_scatter_kernel<<<blocks, 256, 0, stream>>>(feature, src, dst, out, nedges);
    }

    // 3) in-place linear with f32 WMMA: 6250 row-blocks, 10 waves/block
    {
        int nrb = (nnodes + 15) / 16;               // 6250
        int wavesPerBlock = 10;                     // 320 threads
        int blocks = (nrb + wavesPerBlock - 1) / wavesPerBlock;  // 625
        gcn_linear_wmma_kernel<<<blocks, wavesPerBlock * 32, 0, stream>>>(out, W, bias, nrb);
    }
}